// BilinearDecoder_77841987272806
// MI455X (gfx1250) — compile-verified
//
#include <hip/hip_runtime.h>
#include <math.h>

typedef float v2f __attribute__((ext_vector_type(2)));
typedef float v8f __attribute__((ext_vector_type(8)));

#define LATENT 128

// ---------------------------------------------------------------------------
// Kernel 1: u = z @ W   via V_WMMA_F32_16X16X4_F32 (fp32, full precision)
// One wave per 16x16 output tile. 8 waves/block = the 8 column tiles of W.
// Waves are persistent over row tiles so the B panel (128x16 of W, 32 v2f
// fragments = 64 VGPRs) is loaded exactly once per wave.
// ---------------------------------------------------------------------------
__global__ __launch_bounds__(256)
void bilin_gemm_zW(const float* __restrict__ z, const float* __restrict__ W,
                   float* __restrict__ u, int N, int numRowTiles)
{
    const int lane      = threadIdx.x & 31;
    const int waveInBlk = threadIdx.x >> 5;       // 0..7 -> column tile
    const int n0        = waveInBlk * 16;
    const int mrow      = lane & 15;
    const int ncol      = lane & 15;
    const int khalf     = (lane >> 4) << 1;       // 0 (lanes 0-15) or 2 (lanes 16-31)

    // B fragments (ISA 7.12.2, f32 4x16): V0 = K {0|2}, V1 = K {1|3}
    v2f bfrag[32];
#pragma unroll
    for (int kk = 0; kk < 32; ++kk) {
        const int k = kk * 4 + khalf;
        bfrag[kk].x = W[(k + 0) * LATENT + n0 + ncol];
        bfrag[kk].y = W[(k + 1) * LATENT + n0 + ncol];
    }

    for (int rt = blockIdx.x; rt < numRowTiles; rt += gridDim.x) {
        const int m0 = rt << 4;
        int arow = m0 + mrow;
        if (arow >= N) arow = N - 1;              // clamp: EXEC must stay all-ones for WMMA
        const float* zrow = z + (size_t)arow * LATENT + khalf;

        v8f c = {0.f, 0.f, 0.f, 0.f, 0.f, 0.f, 0.f, 0.f};
#pragma unroll
        for (int kk = 0; kk < 32; ++kk) {
            // A fragment (f32 16x4): lane m holds K = khalf, khalf+1 of row m0+m
            v2f a = *(const v2f*)(zrow + kk * 4);
            c = __builtin_amdgcn_wmma_f32_16x16x4_f32(
                    false, a, false, bfrag[kk], (short)0, c, false, false);
        }

        // C layout: VGPR r -> lanes 0-15: (m0+r, n0+lane); lanes 16-31: (m0+8+r, n0+lane-16)
        const int mbase = m0 + ((lane >> 4) << 3);
#pragma unroll
        for (int r = 0; r < 8; ++r) {
            const int m = mbase + r;
            if (m < N) u[(size_t)m * LATENT + n0 + ncol] = c[r];
        }
    }
}

// ---------------------------------------------------------------------------
// Kernel 2: per-edge  out[e] = sigmoid( dot(u[j_e], z[i_e]) )
// One wave32 per edge; lane loads a float4 slice of each 128-float row
// (one global_load_b128 per row per wave; rows are L2-resident: 102 MB < 192 MB).
// ---------------------------------------------------------------------------
__global__ __launch_bounds__(256)
void bilin_edge_dot(const float* __restrict__ z, const float* __restrict__ u,
                    const int* __restrict__ ei, float* __restrict__ out, int E)
{
    const int wave = (int)((blockIdx.x * blockDim.x + threadIdx.x) >> 5);
    const int lane = threadIdx.x & 31;
    if (wave >= E) return;

    const int i = ei[wave];          // edge_index[0][e] -> z_i
    const int j = ei[E + wave];      // edge_index[1][e] -> z_j (goes through W)

    const float4 a = ((const float4*)(z + (size_t)i * LATENT))[lane];
    const float4 b = ((const float4*)(u + (size_t)j * LATENT))[lane];
    float p = a.x * b.x + a.y * b.y + a.z * b.z + a.w * b.w;

#pragma unroll
    for (int off = 16; off > 0; off >>= 1)
        p += __shfl_xor(p, off, 32);

    if (lane == 0) out[wave] = 1.0f / (1.0f + __expf(-p));
}

// ---------------------------------------------------------------------------
// Fallback (only if workspace cannot hold u): direct bilinear per edge.
// ---------------------------------------------------------------------------
__global__ __launch_bounds__(256)
void bilin_edge_direct(const float* __restrict__ z, const float* __restrict__ W,
                       const int* __restrict__ ei, float* __restrict__ out, int E)
{
    const int wave = (int)((blockIdx.x * blockDim.x + threadIdx.x) >> 5);
    const int lane = threadIdx.x & 31;
    if (wave >= E) return;

    const int i = ei[wave];
    const int j = ei[E + wave];

    const float4 zi = ((const float4*)(z + (size_t)i * LATENT))[lane];
    const float* zj = z + (size_t)j * LATENT;

    float4 t = {0.f, 0.f, 0.f, 0.f};
    for (int d = 0; d < LATENT; ++d) {
        const float  s = zj[d];                                   // uniform-ish, cache hit
        const float4 w = ((const float4*)(W + (size_t)d * LATENT))[lane];
        t.x += s * w.x; t.y += s * w.y; t.z += s * w.z; t.w += s * w.w;
    }
    float p = t.x * zi.x + t.y * zi.y + t.z * zi.z + t.w * zi.w;

#pragma unroll
    for (int off = 16; off > 0; off >>= 1)
        p += __shfl_xor(p, off, 32);

    if (lane == 0) out[wave] = 1.0f / (1.0f + __expf(-p));
}

// ---------------------------------------------------------------------------
extern "C" void kernel_launch(void* const* d_in, const int* in_sizes, int n_in,
                              void* d_out, int out_size, void* d_ws, size_t ws_size,
                              hipStream_t stream) {
    const float* z  = (const float*)d_in[0];
    const int*   ei = (const int*)  d_in[1];   // [2, E] int32, row-major
    const float* W  = (const float*)d_in[2];
    float*       out = (float*)d_out;

    const int N = in_sizes[0] / LATENT;
    const int E = in_sizes[1] / 2;

    const size_t uBytes = (size_t)N * LATENT * sizeof(float);
    const int edgeBlocks = (E + 7) / 8;        // 8 waves/block, 1 wave/edge

    if (ws_size >= uBytes) {
        float* u = (float*)d_ws;
        const int numRowTiles = (N + 15) >> 4;
        int gemmBlocks = numRowTiles < 1024 ? numRowTiles : 1024;
        bilin_gemm_zW<<<dim3(gemmBlocks), dim3(256), 0, stream>>>(z, W, u, N, numRowTiles);
        bilin_edge_dot<<<dim3(edgeBlocks), dim3(256), 0, stream>>>(z, u, ei, out, E);
    } else {
        bilin_edge_direct<<<dim3(edgeBlocks), dim3(256), 0, stream>>>(z, W, ei, out, E);
    }
}